// WordHashing_86088324481339
// MI455X (gfx1250) — compile-verified
//
#include <hip/hip_runtime.h>

// ---------------------------------------------------------------------------
// SpMM:  out[16384,300] = relu( sparse[16384,30000] @ W[30000,300] + bias )
// COO sorted by row -> each 16-row tile owns a contiguous nnz segment.
// Accumulate the 16x16xK product on the matrix pipe: V_WMMA_F32_16X16X4_F32.
// ---------------------------------------------------------------------------

typedef __attribute__((ext_vector_type(2))) float v2f;
typedef __attribute__((ext_vector_type(4))) float v4f;
typedef __attribute__((ext_vector_type(8))) float v8f;
typedef __attribute__((ext_vector_type(2))) int   v2i;

#define ROWS_PER_BLOCK 16
#define WAVES          10          // 10 wave32 -> 20 N-tiles of 16 cols >= 300
#define THREADS        (WAVES * 32)
#define CHUNK          128         // nnz staged in LDS per block iteration
#define OUT_DIM        300
#define OUT_PAD        320         // padded LDS row (covers 20 N-tiles)
#define VEC_PER_ROW    (OUT_DIM / 4)   // 75 float4 per output row

__global__ __launch_bounds__(THREADS)
void spmm_wmma_kernel(const int*   __restrict__ sp_rows,
                      const int*   __restrict__ sp_cols,
                      const float* __restrict__ sp_vals,
                      const float* __restrict__ W,      // [30000, 300]
                      const float* __restrict__ bias,   // [300]
                      float*       __restrict__ out,    // [batch, 300]
                      int nnz)
{
    __shared__ int s_range[2];
    __shared__ alignas(16) int   s_row[CHUNK];
    __shared__ alignas(16) int   s_col[CHUNK];
    __shared__ alignas(16) float s_val[CHUNK];
    __shared__ alignas(16) float s_out[ROWS_PER_BLOCK][OUT_PAD];   // 20 KB

    const int tid = threadIdx.x;
    const int r0  = blockIdx.x * ROWS_PER_BLOCK;

    // --- binary search the nnz segment for rows [r0, r0+16) (rows sorted) ---
    if (tid == 0) {
        int lo = 0, hi = nnz;
        while (lo < hi) { int mid = (lo + hi) >> 1;
                          if (sp_rows[mid] < r0) lo = mid + 1; else hi = mid; }
        s_range[0] = lo;
        hi = nnz; const int r1 = r0 + ROWS_PER_BLOCK;
        while (lo < hi) { int mid = (lo + hi) >> 1;
                          if (sp_rows[mid] < r1) lo = mid + 1; else hi = mid; }
        s_range[1] = lo;
    }
    __syncthreads();
    const int segBeg = s_range[0];
    const int segEnd = s_range[1];

    // --- per-lane geometry (wave32; EXEC all-ones around WMMA) ---
    const int wave = tid >> 5;
    const int lane = tid & 31;
    const int half = lane >> 4;        // 0: lanes 0-15, 1: lanes 16-31
    const int ln   = lane & 15;
    const int nb0  = wave * 32 + ln;        // N-tile 0 column
    const int nb1  = wave * 32 + 16 + ln;   // N-tile 1 column
    // clamped 32-bit gather columns (tail cols read col 0; discarded in LDS)
    const unsigned n0 = (unsigned)((nb0 < OUT_DIM) ? nb0 : 0);
    const unsigned n1 = (unsigned)((nb1 < OUT_DIM) ? nb1 : 0);

    v8f c0 = {};   // 16x16 f32 accumulator, tile 0
    v8f c1 = {};   // 16x16 f32 accumulator, tile 1

    for (int base = segBeg; base < segEnd; base += CHUNK) {
        const int cnt = min(CHUNK, segEnd - base);

        // prefetch next chunk of the nnz stream (global_prefetch_b8)
        if (tid == 0 && base + CHUNK < segEnd) {
            __builtin_prefetch(sp_cols + base + CHUNK, 0, 0);
            __builtin_prefetch(sp_vals + base + CHUNK, 0, 0);
        }

        // --- stage CHUNK nnz into LDS; pad with val=0 / row_local=16 ---
        for (int i = tid; i < CHUNK; i += THREADS) {
            const int  g  = base + i;
            const bool ok = (i < cnt);
            s_row[i] = ok ? (sp_rows[g] - r0) : ROWS_PER_BLOCK;
            s_col[i] = ok ? sp_cols[g] : 0;
            s_val[i] = ok ? sp_vals[g] : 0.0f;
        }
        __syncthreads();

        // --- consume in K=4 steps; padded entries are benign, so round the
        //     trip count to a multiple of 8 and unroll x2 (4 WMMAs/trip) ---
        const int cntR = (cnt + 7) & ~7;
        #pragma unroll 2
        for (int j = 0; j < cntR; j += 4) {
            // A-fragment layout (16x4 f32): lane half selects K pair
            const int kA = j + half * 2;           // K = 2*half, K+1 follows

            // paired metadata: kA is even -> 8B-aligned ds_load_b64 each
            const v2i rr = *(const v2i*)&s_row[kA];
            const v2i cc = *(const v2i*)&s_col[kA];
            const v2f vv = *(const v2f*)&s_val[kA];

            v2f a;
            a.x = (rr.x == ln) ? vv.x : 0.0f;      // A[m=ln, kA]
            a.y = (rr.y == ln) ? vv.y : 0.0f;      // A[m=ln, kA+1]

            // B-fragment: 32-bit element offsets -> saddr-form global loads
            const unsigned offA = (unsigned)cc.x * OUT_DIM;
            const unsigned offB = (unsigned)cc.y * OUT_DIM;
            v2f b0, b1;
            b0.x = W[offA + n0]; b0.y = W[offB + n0];
            b1.x = W[offA + n1]; b1.y = W[offB + n1];

            c0 = __builtin_amdgcn_wmma_f32_16x16x4_f32(
                    false, a, false, b0, (short)0, c0, false, false);
            c1 = __builtin_amdgcn_wmma_f32_16x16x4_f32(
                    false, a, false, b1, (short)0, c1, false, false);
        }
        __syncthreads();
    }

    // --- epilogue: dump C tiles to padded LDS (C/D layout: M = v + 8*half,
    //     N = ln), then cooperative coalesced float4 writeback w/ bias+ReLU ---
    #pragma unroll
    for (int v = 0; v < 8; ++v) {
        const int m = v + half * 8;
        s_out[m][nb0] = c0[v];
        s_out[m][nb1] = c1[v];
    }
    __syncthreads();

    const v4f* bias4 = (const v4f*)bias;                 // 75 aligned float4
    for (int i = tid; i < ROWS_PER_BLOCK * VEC_PER_ROW; i += THREADS) {
        const int r  = i / VEC_PER_ROW;
        const int c4 = i - r * VEC_PER_ROW;
        v4f acc = *(const v4f*)&s_out[r][c4 * 4];        // ds_load_b128
        const v4f b = bias4[c4];
        v4f o;
        o.x = fmaxf(acc.x + b.x, 0.0f);
        o.y = fmaxf(acc.y + b.y, 0.0f);
        o.z = fmaxf(acc.z + b.z, 0.0f);
        o.w = fmaxf(acc.w + b.w, 0.0f);
        *(v4f*)&out[(r0 + r) * OUT_DIM + c4 * 4] = o;    // global_store_b128
    }
}

extern "C" void kernel_launch(void* const* d_in, const int* in_sizes, int n_in,
                              void* d_out, int out_size, void* d_ws, size_t ws_size,
                              hipStream_t stream) {
    const int*   sp_rows = (const int*)  d_in[0];
    const int*   sp_cols = (const int*)  d_in[1];
    const float* sp_vals = (const float*)d_in[2];
    const float* weights = (const float*)d_in[3];
    const float* bias    = (const float*)d_in[4];
    float*       out     = (float*)d_out;

    const int nnz   = in_sizes[0];
    const int batch = out_size / OUT_DIM;               // 16384
    const int grid  = (batch + ROWS_PER_BLOCK - 1) / ROWS_PER_BLOCK;

    spmm_wmma_kernel<<<grid, THREADS, 0, stream>>>(
        sp_rows, sp_cols, sp_vals, weights, bias, out, nnz);
}